// IntraModelStructureLoss_14998025798302
// MI455X (gfx1250) — compile-verified
//
#include <hip/hip_runtime.h>
#include <math.h>

typedef float v2f __attribute__((ext_vector_type(2)));
typedef float v8f __attribute__((ext_vector_type(8)));
typedef int   v2i __attribute__((__vector_size__(2 * sizeof(int))));
typedef __attribute__((address_space(1))) v2i gv2i;   // global (prints as __device__)
typedef __attribute__((address_space(3))) v2i lv2i;   // LDS    (prints as __shared__)

#define BN 4096
#define DK 512
// ws layout (floats): c_l@0 q_l@4096 s_l@8192 | c_e@12288 q_e@16384 s_e@20480 | partials@24576
#define WS_CE 12288
#define WS_PART 24576

#if __has_builtin(__builtin_amdgcn_global_load_async_to_lds_b64)
#define HAVE_ASYNC_LDS 1
#else
#define HAVE_ASYNC_LDS 0
#endif

__device__ __forceinline__ float sgnf(float x) {
    return (float)((x > 0.0f) - (x < 0.0f));
}

// ---------------- Kernel 1: per-row norm stats (one wave32 per row, both matrices) ------------
__global__ __launch_bounds__(256) void row_stats_kernel(const float* __restrict__ low,
                                                        const float* __restrict__ emb,
                                                        float* __restrict__ ws) {
    int lane = threadIdx.x & 31;
    int wave = threadIdx.x >> 5;
    int g = blockIdx.x * 8 + wave;          // 0..8191
    int mat = g >> 12;                      // 0 = low, 1 = emb
    int row = g & (BN - 1);
    const float* src = (mat ? emb : low) + (size_t)row * DK;

    float s = 0.0f, ss = 0.0f;
    #pragma unroll
    for (int k = lane; k < DK; k += 32) {
        float v = src[k];
        s += v;
        ss += v * v;
    }
    for (int m = 16; m > 0; m >>= 1) {
        s  += __shfl_xor(s,  m, 32);
        ss += __shfl_xor(ss, m, 32);
    }
    if (lane == 0) {
        float n   = sqrtf(ss);
        float inv = 1.0f / fmaxf(n, 1e-12f);   // EPS_NORM
        float* base = ws + (mat ? WS_CE : 0);
        base[row]            = inv;            // c  : 1/max(norm,eps)
        base[BN + row]       = ss * inv * inv; // q  : ||xn||^2
        base[2 * BN + row]   = s  * inv;       // s  : sum(xn)
    }
}

// ---------------- Kernel 2: fused dual-GEMM distance tiles + ranking-loss epilogue ------------
// 256 blocks x 512 threads (16 waves). Block = one 16-row strip; wave w = column chunk w.
// A panels (low+emb) for the strip staged once into LDS in WMMA-fragment order (64 KB),
// shared by all 16 waves; B panels streamed from global (L2-resident).
__global__ __launch_bounds__(512) void tile_loss_kernel(const float* __restrict__ low,
                                                        const float* __restrict__ emb,
                                                        const float* __restrict__ ws,
                                                        float* __restrict__ partials) {
    const float* c_l = ws;
    const float* q_l = ws + BN;
    const float* s_l = ws + 2 * BN;
    const float* c_e = ws + WS_CE;
    const float* q_e = ws + WS_CE + BN;
    const float* s_e = ws + WS_CE + 2 * BN;

    __shared__ float smem[2 * 128 * 32 * 2];     // 64 KB: [mat][kk][lane] -> 8B fragment
    v2f* shfrag = (v2f*)smem;

    int lane  = threadIdx.x & 31;
    int wave  = threadIdx.x >> 5;               // 0..15 = chunk
    int strip = blockIdx.x;                     // 0..255
    int chunk = wave;
    int task  = strip * 16 + chunk;             // 0..4095
    int i0 = strip * 16;
    int cb = chunk * 256;
    int half = lane >> 4;                       // 0/1
    int lq   = lane & 15;                       // 0..15

    // ---- stage A panels (both matrices) into LDS in fragment order --------------------------
    #pragma unroll 4
    for (int it = 0; it < 16; ++it) {
        int idx = threadIdx.x + 512 * it;       // 0..8191
        int lf  = idx & 31;
        int kk  = (idx >> 5) & 127;
        int mat = idx >> 12;
        const float* src = (mat ? emb : low)
                         + (size_t)(i0 + (lf & 15)) * DK + 4 * kk + 2 * (lf >> 4);
#if HAVE_ASYNC_LDS
        __builtin_amdgcn_global_load_async_to_lds_b64(
            (gv2i*)src, (lv2i*)&shfrag[idx], 0, 0);
#else
        shfrag[idx] = *(const v2f*)src;
#endif
    }
#if HAVE_ASYNC_LDS
#if __has_builtin(__builtin_amdgcn_s_wait_asynccnt)
    __builtin_amdgcn_s_wait_asynccnt(0);
#else
    asm volatile("s_wait_asynccnt 0x0" ::: "memory");
#endif
#endif
    __syncthreads();

    // ---- row-side constants for the 8 C-layout rows this lane owns: i = i0 + r + 8*half -----
    float arow_l[8], crow_l[8], arow_e[8], crow_e[8];
    bool  rowok[8];
    #pragma unroll
    for (int r = 0; r < 8; ++r) {
        int i = i0 + r + 8 * half;
        float ql = q_l[i], sl = s_l[i];
        arow_l[r] = ql + 2e-6f * sl + 5.12e-10f;   // + D*EPS_PD^2 folded into row term
        crow_l[r] = c_l[i];
        float qe = q_e[i], se = s_e[i];
        arow_e[r] = qe + 2e-6f * se + 5.12e-10f;
        crow_e[r] = c_e[i];
        rowok[r] = (i <= BN - 2);                  // rows 0..4094 participate
    }

    const int NT = (chunk == 15) ? 16 : 17;        // 17th (overlap) tile would run off the matrix
    float pend_dl[8], pend_de[8];
    #pragma unroll
    for (int r = 0; r < 8; ++r) { pend_dl[r] = 0.0f; pend_de[r] = 0.0f; }

    float lsum = 0.0f;
    const v2f* shAl = shfrag;                      // [kk*32 + lane]
    const v2f* shAe = shfrag + 128 * 32;

    for (int t = 0; t < NT; ++t) {
        int jb = cb + t * 16;
        const float* pBl = low + (size_t)(jb + lq) * DK + 2 * half;
        const float* pBe = emb + (size_t)(jb + lq) * DK + 2 * half;

        v8f accl = {0.f,0.f,0.f,0.f,0.f,0.f,0.f,0.f};
        v8f acce = {0.f,0.f,0.f,0.f,0.f,0.f,0.f,0.f};

        #pragma unroll 8
        for (int kk = 0; kk < DK / 4; ++kk) {      // 128 steps of K=4
            v2f al = shAl[kk * 32 + lane];         // ds_load_b64, bank-conflict free
            v2f ae = shAe[kk * 32 + lane];
            v2f bl = *(const v2f*)(pBl + 4 * kk);  // global_load_b64, L2-resident
            v2f be = *(const v2f*)(pBe + 4 * kk);
            accl = __builtin_amdgcn_wmma_f32_16x16x4_f32(false, al, false, bl,
                                                         (short)0, accl, false, false);
            acce = __builtin_amdgcn_wmma_f32_16x16x4_f32(false, ae, false, be,
                                                         (short)0, acce, false, false);
        }

        // Column-side terms for this lane's column jc = jb + N  (C layout: N = lane&15)
        int jc = jb + lq;
        float bcol_l = q_l[jc] - 2e-6f * s_l[jc];
        float ccol_l = c_l[jc];
        float bcol_e = q_e[jc] - 2e-6f * s_e[jc];
        float ccol_e = c_e[jc];

        bool inok  = (lq < 15) && (jc < cb + 256) && (jc <= BN - 3);  // in-tile pair j = jc
        bool carry = (t > 0);                                          // boundary pair j = jb-1

        #pragma unroll
        for (int r = 0; r < 8; ++r) {
            float sql = arow_l[r] + bcol_l - 2.0f * accl[r] * crow_l[r] * ccol_l;
            float dl  = sqrtf(fmaxf(sql, 0.0f));
            float sqe = arow_e[r] + bcol_e - 2.0f * acce[r] * crow_e[r] * ccol_e;
            float de  = sqrtf(fmaxf(sqe, 0.0f));

            // next-column values (unconditional shuffles; masked use)
            float dl2 = __shfl_down(dl, 1, 32);
            float de2 = __shfl_down(de, 1, 32);
            if (inok && rowok[r]) {
                float coeff = sgnf(dl - dl2) - sgnf(de - de2);
                lsum += coeff * (de2 - de);
            }
            // carry pair: previous tile's N=15 (lanes 15/31) -> current lanes 0/16
            float pl = __shfl(pend_dl[r], (lane & 16) | 15, 32);
            float pe = __shfl(pend_de[r], (lane & 16) | 15, 32);
            if (carry && (lq == 0) && rowok[r]) {
                float coeff = sgnf(pl - dl) - sgnf(pe - de);
                lsum += coeff * (de - pe);
            }
            pend_dl[r] = dl;
            pend_de[r] = de;
        }
    }

    for (int m = 16; m > 0; m >>= 1) lsum += __shfl_xor(lsum, m, 32);
    if (lane == 0) partials[task] = lsum;
}

// ---------------- Kernel 3: deterministic final reduction ------------------------------------
__global__ __launch_bounds__(256) void reduce_kernel(const float* __restrict__ partials,
                                                     float* __restrict__ out) {
    __shared__ double sh[256];
    int t = threadIdx.x;
    double acc = 0.0;
    for (int j = t * 16; j < t * 16 + 16; ++j) acc += (double)partials[j];
    sh[t] = acc;
    __syncthreads();
    for (int s = 128; s > 0; s >>= 1) {
        if (t < s) sh[t] += sh[t + s];
        __syncthreads();
    }
    if (t == 0) out[0] = (float)(sh[0] / 16764930.0);   // mean over 4095*4094
}

extern "C" void kernel_launch(void* const* d_in, const int* in_sizes, int n_in,
                              void* d_out, int out_size, void* d_ws, size_t ws_size,
                              hipStream_t stream) {
    const float* low = (const float*)d_in[0];   // 4096x512 f32
    const float* emb = (const float*)d_in[1];   // 4096x512 f32
    // d_in[2] (id_numpy) is unused by the reference loss value
    float* out = (float*)d_out;
    float* ws  = (float*)d_ws;
    float* partials = ws + WS_PART;

    row_stats_kernel<<<1024, 256, 0, stream>>>(low, emb, ws);
    tile_loss_kernel<<<256, 512, 0, stream>>>(low, emb, ws, partials);
    reduce_kernel<<<1, 256, 0, stream>>>(partials, out);
}